// MeganEncoder_59871844107147
// MI455X (gfx1250) — compile-verified
//
#include <hip/hip_runtime.h>
#include <hip/hip_bf16.h>
#include <math.h>

// Problem constants (fixed by the reference)
#define NA      100000
#define NE      400000
#define HDIM    256
#define BDIM    128
#define GSEG    2048
#define SSEG    8192
#define NSUB    (SSEG + 1)
#define NHEADS  8
#define ATTD    128
#define HHD     (HDIM / NHEADS)   // 32
#define KBOND   (HDIM + BDIM)     // 384

typedef float v2f __attribute__((ext_vector_type(2)));
typedef float v8f __attribute__((ext_vector_type(8)));

// ---------------------------------------------------------------------------
// LDS-staged fp32 GEMM with bias (+optional ReLU) on the WMMA pipe.
// Y[n,m] = act(X[n,k] @ W[k,m] + b[m]).
// One block = (blockDim.x/32) waves; the block owns one 16-row tile, stages
// A[16,k] in LDS once (coalesced float4 loads), then each wave computes one
// 16x16 output tile, reading A fragments from LDS (ds_load_b64).
// Requires k % 4 == 0 and m % 16 == 0.
// ---------------------------------------------------------------------------
__global__ void gemm_wmma_lds(const float* __restrict__ X,
                              const float* __restrict__ W,
                              const float* __restrict__ B,
                              float* __restrict__ Y,
                              int n, int k, int m, int act) {
    extern __shared__ float As[];            // [16][k]
    const int wpb  = blockDim.x >> 5;
    const int wave = threadIdx.x >> 5;
    const int lane = threadIdx.x & 31;
    const int half = lane >> 4;
    const int l16  = lane & 15;

    const int row0 = blockIdx.x * 16;
    const int col0 = (blockIdx.y * wpb + wave) * 16;

    // --- cooperative coalesced stage of A[16, k] into LDS ---
    {
        const int nvec = (16 * k) >> 2;      // float4 count
        const int kv   = k >> 2;             // float4s per row
        for (int v = threadIdx.x; v < nvec; v += blockDim.x) {
            const int r  = v / kv;
            const int c4 = (v - r * kv) << 2;
            int rr = row0 + r;
            if (rr >= n) rr = n - 1;         // clamp; stores masked later
            const float4 d = *(const float4*)(X + (size_t)rr * k + c4);
            *(float4*)(As + r * k + c4) = d;
        }
    }
    __syncthreads();

    // --- WMMA main loop ---
    const int c = col0 + l16;
    v8f acc = {0.f, 0.f, 0.f, 0.f, 0.f, 0.f, 0.f, 0.f};
    for (int kk = 0; kk < k; kk += 4) {
        const int ka = kk + (half << 1);
        v2f a, bm;
        const float2 af = *(const float2*)(As + l16 * k + ka);   // ds_load_b64
        a.x  = af.x;
        a.y  = af.y;
        bm.x = W[(size_t)ka * m + c];
        bm.y = W[(size_t)(ka + 1) * m + c];
#if defined(__gfx1250__)
        acc = __builtin_amdgcn_wmma_f32_16x16x4_f32(
            /*neg_a=*/false, a, /*neg_b=*/false, bm,
            /*c_mod=*/(short)0, acc, /*reuse_a=*/false, /*reuse_b=*/false);
#endif
    }
    const float bias = B[c];
#pragma unroll
    for (int i = 0; i < 8; ++i) {
        const int rr = row0 + i + 8 * half;  // D layout: VGPR i -> M = i (+8 hi half)
        if (rr < n) {
            float val = acc[i] + bias;
            if (act) val = fmaxf(val, 0.f);
            Y[(size_t)rr * m + c] = val;
        }
    }
}

// ---------------------------------------------------------------------------
// Bond-update GEMM, LDS-staged gathered A rows:
//   A(e, q) = q < HDIM ? out[dst[e]][q] + out[src[e]][q] : bf[e][q - HDIM]
//   Ybf[e, 0:BDIM] = relu(A @ W[KBOND, BDIM] + b)
// One block = 8 waves (256 thr) = all 8 column tiles of one 16-edge tile.
// The gather+add is done once into LDS, then shared by all 8 waves.
// ---------------------------------------------------------------------------
__global__ void bond_gemm_wmma(const float* __restrict__ out,
                               const float* __restrict__ bf,
                               const int*  __restrict__ eidx,
                               const float* __restrict__ W,
                               const float* __restrict__ B,
                               float* __restrict__ Ybf) {
    __shared__ float As[16 * KBOND];         // 24 KB
    const int wave = threadIdx.x >> 5;
    const int lane = threadIdx.x & 31;
    const int half = lane >> 4;
    const int l16  = lane & 15;

    const int row0 = blockIdx.x * 16;
    const int col0 = wave * 16;

    // --- cooperative gathered stage of A[16, KBOND] into LDS ---
    {
        const int nvec = (16 * KBOND) >> 2;  // 1536 float4s
        const int kv   = KBOND >> 2;         // 96 float4s per row
        for (int v = threadIdx.x; v < nvec; v += blockDim.x) {
            const int r  = v / kv;
            const int q4 = (v - r * kv) << 2;
            int e = row0 + r;
            if (e >= NE) e = NE - 1;
            float4 d;
            if (q4 < HDIM) {                 // HDIM % 4 == 0 -> whole vec in-region
                const int s = eidx[2 * e + 0];
                const int dd = eidx[2 * e + 1];
                const float4 a = *(const float4*)(out + (size_t)s  * HDIM + q4);
                const float4 b = *(const float4*)(out + (size_t)dd * HDIM + q4);
                d.x = a.x + b.x; d.y = a.y + b.y; d.z = a.z + b.z; d.w = a.w + b.w;
            } else {
                d = *(const float4*)(bf + (size_t)e * BDIM + (q4 - HDIM));
            }
            *(float4*)(As + r * KBOND + q4) = d;
        }
    }
    __syncthreads();

    const int c = col0 + l16;
    v8f acc = {0.f, 0.f, 0.f, 0.f, 0.f, 0.f, 0.f, 0.f};
    for (int kk = 0; kk < KBOND; kk += 4) {
        const int ka = kk + (half << 1);
        v2f a, bm;
        const float2 af = *(const float2*)(As + l16 * KBOND + ka);
        a.x  = af.x;
        a.y  = af.y;
        bm.x = W[(size_t)ka * BDIM + c];
        bm.y = W[(size_t)(ka + 1) * BDIM + c];
#if defined(__gfx1250__)
        acc = __builtin_amdgcn_wmma_f32_16x16x4_f32(
            false, a, false, bm, (short)0, acc, false, false);
#endif
    }
    const float bias = B[c];
#pragma unroll
    for (int i = 0; i < 8; ++i) {
        const int rr = row0 + i + 8 * half;
        if (rr < NE) {
            Ybf[(size_t)rr * BDIM + c] = fmaxf(acc[i] + bias, 0.f);
        }
    }
}

// ---------------------------------------------------------------------------
// Edge attention logits: logits[e,h] = [xa[dst]|xa[src]|bf[e]] . fatt_W[:,h] + b[h]
// ---------------------------------------------------------------------------
__global__ void fatt_kernel(const float* __restrict__ xa,
                            const float* __restrict__ bf,
                            const int*  __restrict__ eidx,
                            const float* __restrict__ W,
                            const float* __restrict__ B,
                            float* __restrict__ logits) {
    const int t = blockIdx.x * blockDim.x + threadIdx.x;
    if (t >= NE * NHEADS) return;
    const int e = t / NHEADS;
    const int h = t % NHEADS;
    const int s = eidx[2 * e + 0];
    const int d = eidx[2 * e + 1];
    const float* __restrict__ pd = xa + (size_t)d * ATTD;
    const float* __restrict__ ps = xa + (size_t)s * ATTD;
    const float* __restrict__ pb = bf + (size_t)e * BDIM;
    float acc = B[h];
    for (int k = 0; k < ATTD; ++k) acc += pd[k] * W[(size_t)k * NHEADS + h];
    for (int k = 0; k < ATTD; ++k) acc += ps[k] * W[(size_t)(ATTD + k) * NHEADS + h];
    for (int k = 0; k < BDIM; ++k) acc += pb[k] * W[(size_t)(2 * ATTD + k) * NHEADS + h];
    logits[(size_t)e * NHEADS + h] = acc;
}

// ---------------------------------------------------------------------------
// Segment softmax over src (per head) via order-preserving uint max + atomics
// ---------------------------------------------------------------------------
__device__ __forceinline__ unsigned fenc(float f) {
    unsigned u = __float_as_uint(f);
    return (u & 0x80000000u) ? ~u : (u | 0x80000000u);
}
__device__ __forceinline__ float fdec(unsigned u) {
    return (u & 0x80000000u) ? __uint_as_float(u & 0x7FFFFFFFu) : __uint_as_float(~u);
}

__global__ void fill_u32(unsigned* __restrict__ p, unsigned v, int n) {
    int t = blockIdx.x * blockDim.x + threadIdx.x;
    if (t < n) p[t] = v;
}
__global__ void fill_f32(float* __restrict__ p, float v, int n) {
    int t = blockIdx.x * blockDim.x + threadIdx.x;
    if (t < n) p[t] = v;
}

__global__ void segmax_kernel(const float* __restrict__ logits,
                              const int* __restrict__ eidx,
                              unsigned* __restrict__ mx) {
    int t = blockIdx.x * blockDim.x + threadIdx.x;
    if (t >= NE * NHEADS) return;
    int e = t / NHEADS, h = t % NHEADS;
    int s = eidx[2 * e + 0];
    atomicMax(&mx[(size_t)s * NHEADS + h], fenc(logits[t]));
}

__global__ void segexp_kernel(const float* __restrict__ logits,
                              const int* __restrict__ eidx,
                              const unsigned* __restrict__ mx,
                              float* __restrict__ att,
                              float* __restrict__ z) {
    int t = blockIdx.x * blockDim.x + threadIdx.x;
    if (t >= NE * NHEADS) return;
    int e = t / NHEADS, h = t % NHEADS;
    int s = eidx[2 * e + 0];
    float m = fdec(mx[(size_t)s * NHEADS + h]);
    float ex = expf(logits[t] - m);
    att[t] = ex;
    atomicAdd(&z[(size_t)s * NHEADS + h], ex);
}

__global__ void segdiv_kernel(const int* __restrict__ eidx,
                              float* __restrict__ att,
                              const float* __restrict__ z) {
    int t = blockIdx.x * blockDim.x + threadIdx.x;
    if (t >= NE * NHEADS) return;
    int e = t / NHEADS, h = t % NHEADS;
    int s = eidx[2 * e + 0];
    att[t] /= z[(size_t)s * NHEADS + h];
}

// agg[src, c*8 + h] += att[e,h] * v[dst, c]
__global__ void msg_agg_kernel(const float* __restrict__ att,
                               const float* __restrict__ v,
                               const int* __restrict__ eidx,
                               float* __restrict__ agg) {
    int t = blockIdx.x * blockDim.x + threadIdx.x;
    if (t >= NE * HHD) return;
    int e = t / HHD, c = t % HHD;
    int s = eidx[2 * e + 0];
    int d = eidx[2 * e + 1];
    float vd = v[(size_t)d * HHD + c];
    const float* __restrict__ pa = att + (size_t)e * NHEADS;
    float* __restrict__ po = agg + (size_t)s * HDIM + c * NHEADS;
#pragma unroll
    for (int h = 0; h < NHEADS; ++h) atomicAdd(&po[h], pa[h] * vd);
}

// ---------------------------------------------------------------------------
// Masked segment mean scatter + gating
// mask_mode 0: node_mask = (sub != -1 && am == 1);  1: pad_mask = (am == 1)
// ---------------------------------------------------------------------------
__global__ void seg_scatter_feat(const float* __restrict__ feat,
                                 const int* __restrict__ sub,
                                 const int* __restrict__ am,
                                 const int* __restrict__ gid,
                                 float* __restrict__ segfeat,
                                 int use_sub, int mask_mode) {
    int t = blockIdx.x * blockDim.x + threadIdx.x;
    if (t >= NA * HDIM) return;
    int n = t / HDIM, f = t % HDIM;
    int sg = sub[n], a = am[n];
    bool m = (mask_mode == 0) ? (sg != -1 && a == 1) : (a == 1);
    if (!m) return;
    int id = use_sub ? (sg + 1) : gid[n];
    atomicAdd(&segfeat[(size_t)id * HDIM + f], feat[t]);
}

__global__ void seg_scatter_cnt(const int* __restrict__ sub,
                                const int* __restrict__ am,
                                const int* __restrict__ gid,
                                float* __restrict__ segcnt,
                                int use_sub, int mask_mode) {
    int n = blockIdx.x * blockDim.x + threadIdx.x;
    if (n >= NA) return;
    int sg = sub[n], a = am[n];
    bool m = (mask_mode == 0) ? (sg != -1 && a == 1) : (a == 1);
    if (!m) return;
    int id = use_sub ? (sg + 1) : gid[n];
    atomicAdd(&segcnt[id], 1.f);
}

// mode 0 (syn/graph): if node_mask -> out *= sigmoid(mean)
// mode 1 (virt):      if sub == -1 -> out = mean
__global__ void gate_kernel(float* __restrict__ out,
                            const float* __restrict__ segfeat,
                            const float* __restrict__ segcnt,
                            const int* __restrict__ sub,
                            const int* __restrict__ am,
                            const int* __restrict__ gid,
                            int use_sub, int mode) {
    int t = blockIdx.x * blockDim.x + threadIdx.x;
    if (t >= NA * HDIM) return;
    int n = t / HDIM, f = t % HDIM;
    int sg = sub[n], a = am[n];
    int id = use_sub ? (sg + 1) : gid[n];
    float mean = segfeat[(size_t)id * HDIM + f] / fmaxf(segcnt[id], 1.f);
    if (mode == 0) {
        if (sg != -1 && a == 1) out[t] = out[t] * (1.f / (1.f + expf(-mean)));
    } else {
        if (sg == -1) out[t] = mean;
    }
}

// out = relu(out [+ prev])
__global__ void act_kernel(float* __restrict__ out,
                           const float* __restrict__ prev,
                           int do_residual, int n) {
    int t = blockIdx.x * blockDim.x + threadIdx.x;
    if (t >= n) return;
    float v = out[t];
    if (do_residual) v += prev[t];
    out[t] = fmaxf(v, 0.f);
}

// ---------------------------------------------------------------------------
// Host orchestration
// ---------------------------------------------------------------------------
extern "C" void kernel_launch(void* const* d_in, const int* in_sizes, int n_in,
                              void* d_out, int out_size, void* d_ws, size_t ws_size,
                              hipStream_t stream) {
    const float* atom  = (const float*)d_in[0];
    const float* bond  = (const float*)d_in[1];
    const int*   eidx  = (const int*)d_in[2];
    const int*   gid   = (const int*)d_in[3];
    const int*   sgid  = (const int*)d_in[4];
    const int*   amask = (const int*)d_in[5];
    auto P = [&](int li, int idx) -> const float* {
        // per-layer insertion order: att_W,att_b,v_W,v_b,fatt_W,fatt_b,conv_W,conv_b,
        // bond_W,bond_b, syn1_W,syn1_b,syn2_W,syn2_b, graph1..., virt1...
        return (const float*)d_in[6 + li * 22 + idx];
    };

    float* ws = (float*)d_ws;
    size_t o = 0;
    float*    xA      = ws + o; o += (size_t)NA * HDIM;
    float*    xB      = ws + o; o += (size_t)NA * HDIM;
    float*    bfA     = ws + o; o += (size_t)NE * BDIM;
    float*    bfB     = ws + o; o += (size_t)NE * BDIM;
    float*    xa      = ws + o; o += (size_t)NA * ATTD;
    float*    vv      = ws + o; o += (size_t)NA * HHD;
    float*    logits  = ws + o; o += (size_t)NE * NHEADS;
    float*    att     = ws + o; o += (size_t)NE * NHEADS;
    unsigned* mx      = (unsigned*)(ws + o); o += (size_t)NA * NHEADS;
    float*    zz      = ws + o; o += (size_t)NA * NHEADS;
    float*    agg     = ws + o; o += (size_t)NA * HDIM;
    float*    mh      = ws + o; o += (size_t)NA * (HDIM / 2);
    float*    mo      = ws + o; o += (size_t)NA * HDIM;
    float*    segfeat = ws + o; o += (size_t)NSUB * HDIM;
    float*    segcnt  = ws + o; o += (size_t)NSUB;
    (void)ws_size; (void)in_sizes; (void)n_in; (void)out_size;

    auto launch_gemm = [&](const float* X, const float* W, const float* Bp,
                           float* Y, int n, int k, int m, int act) {
        const int tiles = m / 16;
        const int wpb   = (tiles >= 4) ? 4 : tiles;
        dim3 grid((n + 15) / 16, tiles / wpb);
        const size_t shmem = (size_t)16 * k * sizeof(float);
        gemm_wmma_lds<<<grid, 32 * wpb, shmem, stream>>>(X, W, Bp, Y, n, k, m, act);
    };

    struct LayerIO {
        const float* xin; const float* bfin;
        float* xout; float* bfout;
        const float* prev; int residual;
    };
    LayerIO L[4] = {
        { atom, bond, xA, bfA, nullptr, 0 },
        { xA,   bfA,  xB, bfB, atom,    1 },
        { xB,   bfB,  xA, bfA, nullptr, 0 },
        { xA,   bfA,  (float*)d_out, nullptr, xB, 1 },  // last bond update is dead
    };

    const int EH  = NE * NHEADS;
    const int NH  = NA * NHEADS;
    const int NHd = NA * HDIM;

    for (int li = 0; li < 4; ++li) {
        const float* xin  = L[li].xin;
        const float* bfin = L[li].bfin;
        float*       out  = L[li].xout;

        // 1) xa = relu(x @ att_W + att_b)
        launch_gemm(xin, P(li, 0), P(li, 1), xa, NA, HDIM, ATTD, 1);

        // 2) edge attention logits
        fatt_kernel<<<(EH + 255) / 256, 256, 0, stream>>>(
            xa, bfin, eidx, P(li, 4), P(li, 5), logits);

        // 3) segment softmax over src per head
        fill_u32<<<(NH + 255) / 256, 256, 0, stream>>>(mx, 0u, NH);
        segmax_kernel<<<(EH + 255) / 256, 256, 0, stream>>>(logits, eidx, mx);
        fill_f32<<<(NH + 255) / 256, 256, 0, stream>>>(zz, 0.f, NH);
        segexp_kernel<<<(EH + 255) / 256, 256, 0, stream>>>(logits, eidx, mx, att, zz);
        segdiv_kernel<<<(EH + 255) / 256, 256, 0, stream>>>(eidx, att, zz);

        // 4) v = x @ v_W + v_b
        launch_gemm(xin, P(li, 2), P(li, 3), vv, NA, HDIM, HHD, 0);

        // 5) per-head message aggregation into agg[N, H]
        fill_f32<<<(NHd + 255) / 256, 256, 0, stream>>>(agg, 0.f, NHd);
        msg_agg_kernel<<<((NE * HHD) + 255) / 256, 256, 0, stream>>>(att, vv, eidx, agg);

        // 6) out = agg @ conv_W + conv_b
        launch_gemm(agg, P(li, 6), P(li, 7), out, NA, HDIM, HDIM, 0);

        // 7-9) synthon / graph / virtual gating stages
        for (int s = 0; s < 3; ++s) {
            const int pb = 10 + s * 4;
            launch_gemm(out, P(li, pb + 0), P(li, pb + 1), mh, NA, HDIM, HDIM / 2, 1);
            launch_gemm(mh,  P(li, pb + 2), P(li, pb + 3), mo, NA, HDIM / 2, HDIM, 0);
            const int nseg     = (s == 0) ? NSUB : GSEG;
            const int use_sub  = (s == 0) ? 1 : 0;
            const int maskmode = (s == 2) ? 1 : 0;   // virt uses pad_mask
            fill_f32<<<((nseg * HDIM) + 255) / 256, 256, 0, stream>>>(segfeat, 0.f, nseg * HDIM);
            fill_f32<<<(nseg + 255) / 256, 256, 0, stream>>>(segcnt, 0.f, nseg);
            seg_scatter_feat<<<(NHd + 255) / 256, 256, 0, stream>>>(
                mo, sgid, amask, gid, segfeat, use_sub, maskmode);
            seg_scatter_cnt<<<(NA + 255) / 256, 256, 0, stream>>>(
                sgid, amask, gid, segcnt, use_sub, maskmode);
            gate_kernel<<<(NHd + 255) / 256, 256, 0, stream>>>(
                out, segfeat, segcnt, sgid, amask, gid, use_sub, (s == 2) ? 1 : 0);
        }

        // 10) bond update (uses pre-activation out); skipped for final layer
        if (L[li].bfout) {
            bond_gemm_wmma<<<dim3((NE + 15) / 16, 1), 256, 0, stream>>>(
                out, bfin, eidx, P(li, 8), P(li, 9), L[li].bfout);
        }

        // 11) activation / residual
        act_kernel<<<(NHd + 255) / 256, 256, 0, stream>>>(
            out, L[li].prev, L[li].residual, NHd);
    }
}